// MQuantileLoss_23965917511808
// MI455X (gfx1250) — compile-verified
//
#include <hip/hip_runtime.h>
#include <hip/hip_bf16.h>

#define N_COLS      8192
#define B_ROWS      4096
#define TPB         256
#define CHUNK       32               // floats per thread
#define ROW_STRIDE  144              // 128B chunk + 16B skew pad
#define ROW_LDS     (TPB * ROW_STRIDE)   // 36864 bytes

// dynamic LDS layout (dynamic smem starts at LDS offset 0: no static __shared__ here)
#define OFF_PART    36864            // 256 floats  (scan partials)
#define OFF_YB      37888            // 6 floats
#define OFF_YA      37912            // 6 floats
#define OFF_TOT     37936            // 2 floats
#define OFF_LAST    37944            // 2 floats
#define OFF_IDX     37952            // 6 ints
#define SMEM_BYTES  38016

__global__ void mq_rows_kernel(const float* __restrict__ p_est,
                               const float* __restrict__ p_tgt,
                               float* __restrict__ row_out) {
    extern __shared__ char smem[];
    const int t   = threadIdx.x;
    const int row = blockIdx.x;

    float* part  = (float*)(smem + OFF_PART);
    float* sYb   = (float*)(smem + OFF_YB);
    float* sYa   = (float*)(smem + OFF_YA);
    float* sTot  = (float*)(smem + OFF_TOT);
    float* sLast = (float*)(smem + OFF_LAST);
    int*   sIdx  = (int*)  (smem + OFF_IDX);

    const float q0 = 0.25f, q1 = 0.5f, q2 = 0.75f;

    const float* srcs[2] = { p_tgt + (size_t)row * N_COLS,     // phase 0: target
                             p_est + (size_t)row * N_COLS };   // phase 1: estimate

    for (int p = 0; p < 2; ++p) {
        // ---- async stream row -> LDS (CDNA5 GLOBAL_LOAD_ASYNC_TO_LDS, ASYNCcnt) ----
        const char* g = (const char*)srcs[p];
        #pragma unroll
        for (int it = 0; it < 8; ++it) {
            const int goff  = it * 4096 + t * 16;          // coalesced 16B per lane
            const int chunk = it * 32 + (t >> 3);          // 128B chunk index
            const unsigned loff = (unsigned)(goff + chunk * 16);  // bank-skew pad
            const char* ga = g + goff;
            asm volatile("global_load_async_to_lds_b128 %0, %1, off"
                         :: "v"(loff), "v"(ga) : "memory");
        }
        asm volatile("s_wait_asynccnt 0x0" ::: "memory");
        __syncthreads();

        // ---- pull my 32-float chunk into registers, chunk sum ----
        float4 r[8];
        const float4* my = (const float4*)(smem + t * ROW_STRIDE);
        float s = 0.f;
        #pragma unroll
        for (int i = 0; i < 8; ++i) {
            r[i] = my[i];
            s += r[i].x + r[i].y + r[i].z + r[i].w;
        }
        part[t] = s;
        if (t < 3)       sIdx[p * 3 + t] = -1;
        if (t == TPB - 1) sLast[p] = r[7].w;               // row's last element
        __syncthreads();

        // ---- Hillis-Steele inclusive scan of 256 chunk sums ----
        #pragma unroll
        for (int off = 1; off < TPB; off <<= 1) {
            float v = part[t];
            float a = (t >= off) ? part[t - off] : 0.f;
            __syncthreads();
            part[t] = v + a;
            __syncthreads();
        }
        const float excl = (t > 0) ? part[t - 1] : 0.f;
        if (t == 0) sTot[p] = part[TPB - 1];

        // ---- walk my chunk's running CDF, record quantile crossings ----
        float run = excl;
        const int base = t * CHUNK;
        #pragma unroll
        for (int i = 0; i < 8; ++i) {
            const float xs[4] = { r[i].x, r[i].y, r[i].z, r[i].w };
            #pragma unroll
            for (int j = 0; j < 4; ++j) {
                const float prev = run;
                run = prev + xs[j];
                const int gi = base + i * 4 + j;
                if (prev < q0 && run >= q0) { sIdx[p*3+0] = gi; sYb[p*3+0] = run; sYa[p*3+0] = prev; }
                if (prev < q1 && run >= q1) { sIdx[p*3+1] = gi; sYb[p*3+1] = run; sYa[p*3+1] = prev; }
                if (prev < q2 && run >= q2) { sIdx[p*3+2] = gi; sYb[p*3+2] = run; sYa[p*3+2] = prev; }
            }
        }
        __syncthreads();   // results visible; row buffer safe to overwrite next phase
    }

    // ---- interpolate + per-row |diff| sum ----
    if (t == 0) {
        const float qs[3] = { q0, q1, q2 };
        float sc[2][3];
        #pragma unroll
        for (int p = 0; p < 2; ++p) {
            #pragma unroll
            for (int k = 0; k < 3; ++k) {
                int i = sIdx[p * 3 + k];
                float yb, ya;
                if (i < 0) {  // q never reached: clip to n-1 (reference's jnp.clip path)
                    i  = N_COLS - 1;
                    yb = sTot[p];
                    ya = sTot[p] - sLast[p];
                } else {
                    yb = sYb[p * 3 + k];
                    ya = sYa[p * 3 + k];
                }
                sc[p][k] = (float)(i + 1) + (qs[k] - yb) / (yb - ya);
            }
        }
        row_out[row] = fabsf(sc[0][0] - sc[1][0])
                     + fabsf(sc[0][1] - sc[1][1])
                     + fabsf(sc[0][2] - sc[1][2]);
    }
}

__global__ void mq_reduce_kernel(const float* __restrict__ row_out,
                                 float* __restrict__ out) {
    __shared__ float red[256];
    const int t = threadIdx.x;
    float s = 0.f;
    for (int i = t; i < B_ROWS; i += 256) s += row_out[i];
    red[t] = s;
    __syncthreads();
    for (int off = 128; off > 0; off >>= 1) {
        if (t < off) red[t] += red[t + off];
        __syncthreads();
    }
    if (t == 0) out[0] = red[0] / (float)(B_ROWS * 3);
}

extern "C" void kernel_launch(void* const* d_in, const int* in_sizes, int n_in,
                              void* d_out, int out_size, void* d_ws, size_t ws_size,
                              hipStream_t stream) {
    // setup_inputs order: {"p_estimate", "p_target"} — both fp32 [4096, 8192]
    const float* p_est = (const float*)d_in[0];
    const float* p_tgt = (const float*)d_in[1];
    float* row_partials = (float*)d_ws;            // 4096 floats of scratch
    float* out = (float*)d_out;

    mq_rows_kernel<<<B_ROWS, TPB, SMEM_BYTES, stream>>>(p_est, p_tgt, row_partials);
    mq_reduce_kernel<<<1, 256, 0, stream>>>(row_partials, out);
}